// ThresholdSegmentation_30459908063386
// MI455X (gfx1250) — compile-verified
//
#include <hip/hip_runtime.h>
#include <stdint.h>

// Problem geometry
constexpr int B_IMG   = 64;
constexpr int PLANE   = 512 * 512;          // 262144 pixels / image
constexpr int NBINS   = 256;
constexpr int BPI     = 32;                 // histogram blocks per image
constexpr int SEG     = PLANE / BPI;        // 8192 pixels per histogram block
constexpr int TPB     = 256;                // threads per block
constexpr int TILE_PX = TPB * 4;            // 1024 px per tile (4 px / thread)
constexpr int ITERS   = SEG / TILE_PX;      // 8 tiles per block
constexpr int MBPI    = PLANE / (256 * 4);  // 256 mask blocks per image

// Async copy: one b128 per lane into LDS, tracked by ASYNCcnt (CDNA5 path).
// GVS addressing: saddr = uniform 64-bit base, vaddr = per-lane 32-bit byte
// offset, channel plane stride via the signed-24-bit instruction offset.
#define ASYNC_LOAD_B128(ldsoff, goff, base, imm)                               \
    asm volatile("global_load_async_to_lds_b128 %0, %1, %2 offset:" #imm       \
                 :: "v"(ldsoff), "v"(goff), "s"(base) : "memory")

__device__ __forceinline__ void wait_asynccnt_le3() {
    asm volatile("s_wait_asynccnt 0x3" ::: "memory");
}
__device__ __forceinline__ void wait_asynccnt_0() {
    asm volatile("s_wait_asynccnt 0x0" ::: "memory");
}

// ---------------------------------------------------------------- gray helper
__device__ __forceinline__ int gray_of(float r, float g, float b) {
    // u8 = clip(floor(x*255), 0, 255)   (exact fp32, no contraction)
    float ur = fminf(fmaxf(floorf(__fmul_rn(r, 255.0f)), 0.0f), 255.0f);
    float ug = fminf(fmaxf(floorf(__fmul_rn(g, 255.0f)), 0.0f), 255.0f);
    float ub = fminf(fmaxf(floorf(__fmul_rn(b, 255.0f)), 0.0f), 255.0f);
    // gray = clip(rint(0.299*r + 0.587*g + 0.114*b), 0, 255); left-assoc like jnp
    float s  = __fadd_rn(__fadd_rn(__fmul_rn(0.299f, ur), __fmul_rn(0.587f, ug)),
                         __fmul_rn(0.114f, ub));
    return (int)fminf(fmaxf(rintf(s), 0.0f), 255.0f);
}

// ---------------------------------------------------------------- zero hist
__global__ void __launch_bounds__(256) zero_hist_kernel(unsigned* __restrict__ hist) {
    hist[blockIdx.x * 256 + threadIdx.x] = 0u;   // 64 blocks x 256 = 64*NBINS
}

// ---------------------------------------------------------------- histogram
// Double-buffered async-to-LDS pipeline: while tile i is consumed (ds_load ->
// gray math -> ds_add privatized histograms), tile i+1 streams into the other
// LDS buffer via ASYNCcnt. Each thread consumes exactly the 16B/channel chunk
// it requested, so no inter-thread barrier is needed inside the pipeline.
template <bool STORE_GRAY>
__global__ void __launch_bounds__(256) hist_kernel(const float* __restrict__ x,
                                                   unsigned* __restrict__ hist,
                                                   uint8_t* __restrict__ grayc) {
    __shared__ float4   abuf[2][3 * TPB];        // 2 x 12KB RGB tiles
    __shared__ unsigned lh[8 * NBINS];           // one sub-hist per wave32

    const int tid  = threadIdx.x;
    const int wave = tid >> 5;                   // wave32
    const int img  = blockIdx.x / BPI;
    const int blk  = blockIdx.x % BPI;

    #pragma unroll
    for (int j = 0; j < 8; ++j) lh[j * NBINS + tid] = 0u;
    __syncthreads();

    const float* base = x + (size_t)img * 3 * PLANE + (size_t)blk * SEG;  // R plane
    unsigned* gout = STORE_GRAY
        ? (unsigned*)(grayc + (size_t)img * PLANE + (size_t)blk * SEG) : nullptr;

    // LDS byte offsets of this thread's chunks (low 32 bits of flat shared addr)
    unsigned lR[2], lG[2], lB[2];
    #pragma unroll
    for (int s = 0; s < 2; ++s) {
        lR[s] = (unsigned)(uintptr_t)&abuf[s][tid];
        lG[s] = (unsigned)(uintptr_t)&abuf[s][TPB + tid];
        lB[s] = (unsigned)(uintptr_t)&abuf[s][2 * TPB + tid];
    }

    auto issue = [&](int it, int sel) {
        const unsigned goff = (unsigned)((it * TILE_PX + tid * 4) * sizeof(float));
        ASYNC_LOAD_B128(lR[sel], goff, base, 0);        // R plane
        ASYNC_LOAD_B128(lG[sel], goff, base, 1048576);  // +PLANE   floats (1 MB)
        ASYNC_LOAD_B128(lB[sel], goff, base, 2097152);  // +2*PLANE floats (2 MB)
    };

    issue(0, 0);
    for (int it = 0; it < ITERS; ++it) {
        const int cur = it & 1;
        if (it + 1 < ITERS) {
            issue(it + 1, cur ^ 1);
            wait_asynccnt_le3();   // next tile's 3 in flight; current tile landed
        } else {
            wait_asynccnt_0();
        }

        const float4 r4 = abuf[cur][tid];
        const float4 g4 = abuf[cur][TPB + tid];
        const float4 b4 = abuf[cur][2 * TPB + tid];

        int q0 = gray_of(r4.x, g4.x, b4.x);
        int q1 = gray_of(r4.y, g4.y, b4.y);
        int q2 = gray_of(r4.z, g4.z, b4.z);
        int q3 = gray_of(r4.w, g4.w, b4.w);

        atomicAdd(&lh[wave * NBINS + q0], 1u);   // ds_add_u32, privatized per wave
        atomicAdd(&lh[wave * NBINS + q1], 1u);
        atomicAdd(&lh[wave * NBINS + q2], 1u);
        atomicAdd(&lh[wave * NBINS + q3], 1u);

        if (STORE_GRAY) {
            unsigned pk = (unsigned)q0 | ((unsigned)q1 << 8) |
                          ((unsigned)q2 << 16) | ((unsigned)q3 << 24);
            gout[(it * TILE_PX + tid * 4) >> 2] = pk;
        }
    }
    __syncthreads();

    unsigned sum = 0;
    #pragma unroll
    for (int j = 0; j < 8; ++j) sum += lh[j * NBINS + tid];
    if (sum) atomicAdd(&hist[img * NBINS + tid], sum);
}

// ---------------------------------------------------------------- Otsu (tiny)
__global__ void __launch_bounds__(64) otsu_kernel(const unsigned* __restrict__ hist,
                                                  int* __restrict__ thresh) {
    const int b = threadIdx.x;
    if (b >= B_IMG) return;
    const unsigned* h = hist + b * NBINS;
    const float invN = 1.0f / 262144.0f;         // 2^-18: p = hist*invN is EXACT

    float mu = 0.0f;                              // pass 1: mu_t = cumsum(bins*p)[-1]
    for (int k = 0; k < NBINS; ++k) {
        float p = __fmul_rn((float)h[k], invN);
        mu = __fadd_rn(mu, __fmul_rn((float)k, p));
    }
    const float mu_t = mu;

    float omega = 0.0f; mu = 0.0f;
    float best = -3.402823466e38f; int bestk = 0; // first-occurrence argmax
    for (int k = 0; k < NBINS; ++k) {
        float p = __fmul_rn((float)h[k], invN);
        omega = __fadd_rn(omega, p);
        mu    = __fadd_rn(mu, __fmul_rn((float)k, p));
        float denom = __fmul_rn(omega, __fsub_rn(1.0f, omega));
        float s;
        if (denom > 1e-12f) {
            float d = __fsub_rn(__fmul_rn(mu_t, omega), mu);
            s = __fmul_rn(d, d) / fmaxf(denom, 1e-12f);
        } else {
            s = -1.0f;
        }
        if (s > best) { best = s; bestk = k; }
    }
    thresh[b] = bestk;
}

// ---------------------------------------------------------------- mask (cached gray)
__global__ void __launch_bounds__(256) mask_from_gray_kernel(const uint8_t* __restrict__ grayc,
                                                             const int* __restrict__ thresh,
                                                             float* __restrict__ out) {
    const int img = blockIdx.x / MBPI;
    const int blk = blockIdx.x % MBPI;
    const int t   = thresh[img];
    const int base = blk * 1024 + threadIdx.x * 4;           // pixel index in image
    const unsigned g4 = ((const unsigned*)(grayc + (size_t)img * PLANE))[base >> 2];
    float4 o;
    o.x = ((int)( g4        & 0xFF) > t) ? 1.0f : 0.0f;
    o.y = ((int)((g4 >>  8) & 0xFF) > t) ? 1.0f : 0.0f;
    o.z = ((int)((g4 >> 16) & 0xFF) > t) ? 1.0f : 0.0f;
    o.w = ((int)((g4 >> 24) & 0xFF) > t) ? 1.0f : 0.0f;
    *(float4*)(out + (size_t)img * PLANE + base) = o;
}

// ---------------------------------------------------------------- mask (recompute, ws fallback)
__global__ void __launch_bounds__(256) mask_recompute_kernel(const float* __restrict__ x,
                                                             const int* __restrict__ thresh,
                                                             float* __restrict__ out) {
    const int img = blockIdx.x / MBPI;
    const int blk = blockIdx.x % MBPI;
    const int t   = thresh[img];
    const int base = blk * 1024 + threadIdx.x * 4;
    const float* rp = x + (size_t)img * 3 * PLANE + base;
    const float4 r4 = *(const float4*)(rp);
    const float4 g4 = *(const float4*)(rp + PLANE);
    const float4 b4 = *(const float4*)(rp + 2 * PLANE);
    float4 o;
    o.x = (gray_of(r4.x, g4.x, b4.x) > t) ? 1.0f : 0.0f;
    o.y = (gray_of(r4.y, g4.y, b4.y) > t) ? 1.0f : 0.0f;
    o.z = (gray_of(r4.z, g4.z, b4.z) > t) ? 1.0f : 0.0f;
    o.w = (gray_of(r4.w, g4.w, b4.w) > t) ? 1.0f : 0.0f;
    *(float4*)(out + (size_t)img * PLANE + base) = o;
}

// ---------------------------------------------------------------- launch
extern "C" void kernel_launch(void* const* d_in, const int* in_sizes, int n_in,
                              void* d_out, int out_size, void* d_ws, size_t ws_size,
                              hipStream_t stream) {
    const float* x = (const float*)d_in[0];
    float* out = (float*)d_out;

    uint8_t* ws     = (uint8_t*)d_ws;
    unsigned* hist  = (unsigned*)ws;                              // 64*256*4 = 64 KB
    int* thresh     = (int*)(ws + B_IMG * NBINS * sizeof(unsigned));
    uint8_t* grayc  = ws + B_IMG * NBINS * sizeof(unsigned) + 256;
    const size_t need = (size_t)B_IMG * NBINS * sizeof(unsigned) + 256 +
                        (size_t)B_IMG * PLANE;                    // ~16.8 MB
    const bool cache = (ws_size >= need);                         // deterministic branch

    zero_hist_kernel<<<B_IMG, 256, 0, stream>>>(hist);

    if (cache) {
        hist_kernel<true><<<B_IMG * BPI, 256, 0, stream>>>(x, hist, grayc);
    } else {
        hist_kernel<false><<<B_IMG * BPI, 256, 0, stream>>>(x, hist, nullptr);
    }

    otsu_kernel<<<1, 64, 0, stream>>>(hist, thresh);

    if (cache) {
        mask_from_gray_kernel<<<B_IMG * MBPI, 256, 0, stream>>>(grayc, thresh, out);
    } else {
        mask_recompute_kernel<<<B_IMG * MBPI, 256, 0, stream>>>(x, thresh, out);
    }
}